// SCM_32856499814807
// MI455X (gfx1250) — compile-verified
//
#include <hip/hip_runtime.h>

#define NROWS 4096
#define DD    128
#define H1D   256
#define NF    (NROWS * DD)

typedef __attribute__((ext_vector_type(2))) float v2f;
typedef __attribute__((ext_vector_type(8))) float v8f;

// ---------------------------------------------------------------------------
// 128x128 matmul C = A*B over LDS buffers using f32 WMMA 16x16x4.
// 8 waves, 8 output tiles each.  Lane layout (ISA 7.12.2, wave32):
//   half = lane>>4, li = lane&15
//   A(16x4):  a[q] = A[m=li][k = 2*half + q]
//   B(4x16):  b[q] = B[k = 2*half + q][n=li]
//   C(16x16): c[r] = C[m = r + 8*half][n=li]
// ---------------------------------------------------------------------------
__device__ __forceinline__ void mm128(const float* __restrict__ A,
                                      const float* __restrict__ B,
                                      float* __restrict__ C,
                                      int wave, int lane) {
  const int half = lane >> 4, li = lane & 15;
  for (int tdx = wave; tdx < 64; tdx += 8) {
    const int mt = (tdx >> 3) * 16;
    const int nt = (tdx & 7) * 16;
    v8f c = {};
    #pragma unroll 4
    for (int kk = 0; kk < DD; kk += 4) {
      const int k0 = kk + 2 * half;
      v2f a, b;
      a.x = A[(mt + li) * DD + k0];
      a.y = A[(mt + li) * DD + k0 + 1];
      b.x = B[k0 * DD + nt + li];
      b.y = B[(k0 + 1) * DD + nt + li];
      c = __builtin_amdgcn_wmma_f32_16x16x4_f32(false, a, false, b,
                                                (short)0, c, false, false);
    }
    #pragma unroll
    for (int r = 0; r < 8; ++r)
      C[(mt + r + 8 * half) * DD + nt + li] = c[r];
  }
}

// ---------------------------------------------------------------------------
// Single fused kernel, grid = 257 blocks x 256 threads.
//
// Blocks 0..255 (GEMM+MLP path), block b owns rows [16b, 16b+16):
//   Wl = weight*A staged into LDS; w2s[j]=sum_k W2[j,k], b1 into LDS.
//   y-tile = x@W + bias via f32 WMMA (wave w -> column tile w), then
//   f[n,i] = sum_j relu(y*W1[i,j]+b1[j])*w2s[j] + sum(b2).
//
// Block 256 (h_val + reg path):
//   M = I + (weight*A)^2/D built in LDS; reg = sum|W| reduced;
//   E = M^127 by repeated squaring (P=M,R=M; 6x {T=P*P; R=R*T; P=T});
//   h = sum_ij E[j,i]*M[i,j] - D  (M recomputed elementwise from global).
// ---------------------------------------------------------------------------
__global__ __launch_bounds__(256) void fused_kernel(
    const float* __restrict__ x, const float* __restrict__ weight,
    const float* __restrict__ bias, const float* __restrict__ A,
    const float* __restrict__ W1, const float* __restrict__ b1,
    const float* __restrict__ W2, const float* __restrict__ b2,
    float* __restrict__ fout, float* __restrict__ hout,
    float* __restrict__ regout) {
  extern __shared__ float smem[];
  const int t = threadIdx.x;
  const int wave = t >> 5;
  const int lane = t & 31;
  const int half = lane >> 4;
  const int li   = lane & 15;

  if (blockIdx.x < NROWS / 16) {
    // ---------------- GEMM + fused MLP path ----------------
    float* Wl  = smem;               // 128*128
    float* b1l = smem + DD * DD;     // 256
    float* w2l = b1l + H1D;          // 256

    for (int idx = t; idx < DD * DD; idx += 256)
      Wl[idx] = weight[idx] * A[idx];
    {
      float s = 0.f;
      for (int k = 0; k < DD; ++k) s += W2[t * DD + k];   // W2 row sums
      w2l[t] = s;
      b1l[t] = b1[t];
    }
    float bb = 0.f;                                       // sum(b2), uniform
    for (int k = 0; k < DD; ++k) bb += b2[k];
    __syncthreads();

    const int rowbase = blockIdx.x * 16;
    const int col = wave * 16 + li;                        // i in [0,128)

    v8f c = {};
    const float* xrow = x + (size_t)(rowbase + li) * DD;
    #pragma unroll 4
    for (int kk = 0; kk < DD; kk += 4) {
      const int k0 = kk + 2 * half;
      v2f a, b;
      a.x = xrow[k0];
      a.y = xrow[k0 + 1];
      b.x = Wl[k0 * DD + col];
      b.y = Wl[(k0 + 1) * DD + col];
      c = __builtin_amdgcn_wmma_f32_16x16x4_f32(false, a, false, b,
                                                (short)0, c, false, false);
    }

    const float bcol = bias[col];
    float yv[8], acc[8];
    #pragma unroll
    for (int r = 0; r < 8; ++r) { yv[r] = c[r] + bcol; acc[r] = 0.f; }

    const float* W1row = W1 + (size_t)col * H1D;
    for (int j = 0; j < H1D; ++j) {
      const float w1 = W1row[j];
      const float bj = b1l[j];
      const float wj = w2l[j];
      #pragma unroll
      for (int r = 0; r < 8; ++r) {
        float h = fmaf(yv[r], w1, bj);
        h = fmaxf(h, 0.f);
        acc[r] = fmaf(h, wj, acc[r]);
      }
    }
    #pragma unroll
    for (int r = 0; r < 8; ++r) {
      const int n = rowbase + r + 8 * half;
      fout[(size_t)n * DD + col] = acc[r] + bb;
    }
  } else {
    // ---------------- h_val + reg path (one block) ----------------
    float* buf0 = smem;
    float* buf1 = smem + DD * DD;
    float* buf2 = smem + 2 * DD * DD;
    float* red  = smem + 3 * DD * DD;   // 256 floats

    float regsum = 0.f;
    for (int idx = t; idx < DD * DD; idx += 256) {
      const float w = weight[idx] * A[idx];
      regsum += fabsf(w);
      const int i = idx / DD, j = idx % DD;
      const float m = fmaf(w * w, 1.0f / (float)DD, (i == j) ? 1.0f : 0.0f);
      buf0[idx] = m;   // P = M^1
      buf1[idx] = m;   // R = M^1
    }
    red[t] = regsum;
    __syncthreads();
    for (int s = 128; s > 0; s >>= 1) {
      if (t < s) red[t] += red[t + s];
      __syncthreads();
    }
    if (t == 0) regout[0] = red[0];
    __syncthreads();

    float* P = buf0;
    float* R = buf1;
    float* T = buf2;
    for (int it = 0; it < 6; ++it) {
      mm128(P, P, T, wave, lane);   // T = P^2
      __syncthreads();
      mm128(R, T, P, wave, lane);   // old-P buffer <- R * P^2  (new R)
      __syncthreads();
      float* newP = T;              // rotate buffers
      float* newR = P;
      T = R; P = newP; R = newR;
    }
    // R now holds E = M^127; recompute M elementwise for (E^T ⊙ M).sum()
    float s = 0.f;
    for (int idx = t; idx < DD * DD; idx += 256) {
      const int i = idx / DD, j = idx % DD;
      const float w = weight[idx] * A[idx];
      const float m = fmaf(w * w, 1.0f / (float)DD, (i == j) ? 1.0f : 0.0f);
      s += R[j * DD + i] * m;
    }
    red[t] = s;
    __syncthreads();
    for (int st = 128; st > 0; st >>= 1) {
      if (t < st) red[t] += red[t + st];
      __syncthreads();
    }
    if (t == 0) hout[0] = red[0] - (float)DD;
  }
}

// ---------------------------------------------------------------------------
extern "C" void kernel_launch(void* const* d_in, const int* in_sizes, int n_in,
                              void* d_out, int out_size, void* d_ws, size_t ws_size,
                              hipStream_t stream) {
  const float* x      = (const float*)d_in[0];
  const float* weight = (const float*)d_in[1];
  const float* bias   = (const float*)d_in[2];
  const float* A      = (const float*)d_in[3];
  const float* W1     = (const float*)d_in[4];
  const float* b1     = (const float*)d_in[5];
  const float* W2     = (const float*)d_in[6];
  const float* b2     = (const float*)d_in[7];

  float* fout   = (float*)d_out;        // [4096,128]
  float* hout   = fout + NF;            // scalar h_val
  float* regout = fout + NF + 1;        // scalar reg

  // LDS: max(GEMM path 67KB, h_val path 3*64KB + 1KB) -> one uniform request
  const size_t smemBytes = (size_t)(3 * DD * DD + 256) * sizeof(float); // ~193KB
  fused_kernel<<<NROWS / 16 + 1, 256, smemBytes, stream>>>(
      x, weight, bias, A, W1, b1, W2, b2, fout, hout, regout);
}